// Repeat_53111565582514
// MI455X (gfx1250) — compile-verified
//
#include <hip/hip_runtime.h>

// Leave-one-out gather:
//   out[i, j, :, :] = patches[j + (j >= i), :, :]
// patches: [256, 8, 768] f32  (6 MB, L2-resident)
// out:     [256, 255, 8, 768] f32 (1.604 GB, streaming writes)
//
// Pure bandwidth kernel: ~69 us floor at 23.3 TB/s. Strategy:
//  - one 256-thread block (8 wave32) per 24 KB output row
//  - 128-bit vector loads (RT, reuse from L2) and 128-bit NT stores
//  - grid (255, 256): j = blockIdx.x, i = blockIdx.y -> no integer division

typedef float v4f __attribute__((ext_vector_type(4)));

#define PATCH_NUM   256
#define ROW_FLOATS  6144            // 8 * 768
#define ROW_VEC4    (ROW_FLOATS/4)  // 1536 float4 per row
#define BLOCK_THREADS 256
#define VEC_PER_THREAD (ROW_VEC4 / BLOCK_THREADS)  // 6

__global__ __launch_bounds__(BLOCK_THREADS)
void loo_gather_kernel(const v4f* __restrict__ src, v4f* __restrict__ dst) {
    const int j = blockIdx.x;                 // 0..254
    const int i = blockIdx.y;                 // 0..255
    const int srcRow = j + (j >= i ? 1 : 0);  // leave-one-out index

    const v4f* __restrict__ s = src + (size_t)srcRow * ROW_VEC4;
    v4f* __restrict__       d = dst + ((size_t)i * (PATCH_NUM - 1) + (size_t)j) * ROW_VEC4;

    const int t = threadIdx.x;

    // Warm the source row into cache (global_prefetch_b8 on gfx1250).
    __builtin_prefetch((const void*)(s + t), 0, 3);

#pragma unroll
    for (int k = 0; k < VEC_PER_THREAD; ++k) {
        const int idx = t + k * BLOCK_THREADS;       // fully coalesced within wave
        v4f v = s[idx];                              // RT load: L2 hit after first touch
        __builtin_nontemporal_store(v, &d[idx]);     // NT 128-bit streaming store
    }
}

extern "C" void kernel_launch(void* const* d_in, const int* in_sizes, int n_in,
                              void* d_out, int out_size, void* d_ws, size_t ws_size,
                              hipStream_t stream) {
    const v4f* src = (const v4f*)d_in[0];   // patches, float32
    v4f* dst = (v4f*)d_out;                 // float32 output

    dim3 grid(PATCH_NUM - 1, PATCH_NUM);    // (j, i) = 255 x 256 = 65,280 row-copies
    loo_gather_kernel<<<grid, BLOCK_THREADS, 0, stream>>>(src, dst);
}